// VQALoss_34359738695
// MI455X (gfx1250) — compile-verified
//
#include <hip/hip_runtime.h>
#include <stdint.h>

// Problem constants (from reference)
#define BATCH  16384
#define CLS    3129
#define NA     10
#define TPB    256
#define NWAVES (TPB / 32)
// 782 * 16B = 12512 B = 3128 floats via b128 async chunks; 1 tail float via b32
#define CHUNKS 782

// Wave32 butterfly reductions (fixed pattern -> deterministic)
__device__ inline float waveMax(float v) {
#pragma unroll
    for (int off = 16; off > 0; off >>= 1) v = fmaxf(v, __shfl_xor(v, off, 32));
    return v;
}
__device__ inline float waveSum(float v) {
#pragma unroll
    for (int off = 16; off > 0; off >>= 1) v += __shfl_xor(v, off, 32);
    return v;
}

// --------------------------------------------------------------------------
// Kernel 1: one workgroup per row.
//  - DMA the row into LDS with gfx1250 async global->LDS loads, non-temporal
//    (205 MB stream > 192 MB L2: read once, don't thrash)
//  - two-pass stable log-softmax out of LDS (max, then sum of exp)
//  - 10 soft answers gathered in parallel; emits float4 partial
//    {weighted_i, w_y*nll_i, w_y, (y==0)}
// --------------------------------------------------------------------------
__global__ __launch_bounds__(TPB) void vqa_row_kernel(
    const float* __restrict__ pred,
    const float* __restrict__ weight,
    const int*   __restrict__ answers,
    const int*   __restrict__ mode_answer,
    float4*      __restrict__ partials)
{
    __shared__ float row[CLS];
    __shared__ float wred[NWAVES];
    __shared__ float tacc[NA];

    const int tid  = threadIdx.x;
    const int lane = tid & 31;
    const int wave = tid >> 5;
    const int i    = blockIdx.x;

    // Uniform SGPR base for this row; per-lane 32-bit byte offsets (GVS mode).
    const uint64_t base    = (uint64_t)(uintptr_t)(pred + (size_t)i * CLS);
    // Low 32 bits of a generic LDS pointer == byte offset in the workgroup's LDS.
    const uint32_t rowBase = (uint32_t)(uintptr_t)row;

    // Async DMA: global -> LDS, 16B per lane per issue. Tracked by ASYNCcnt.
    for (int c = tid; c < CHUNKS; c += TPB) {
        uint32_t voff  = (uint32_t)c * 16u;
        uint32_t laddr = rowBase + voff;
        asm volatile("global_load_async_to_lds_b128 %0, %1, %2 th:TH_LOAD_NT"
                     :: "v"(laddr), "v"(voff), "s"(base) : "memory");
    }
    if (tid == 0) {  // tail element 3128
        uint32_t voff  = (uint32_t)CHUNKS * 16u;
        uint32_t laddr = rowBase + voff;
        asm volatile("global_load_async_to_lds_b32 %0, %1, %2 th:TH_LOAD_NT"
                     :: "v"(laddr), "v"(voff), "s"(base) : "memory");
    }
    asm volatile("s_wait_asynccnt 0" ::: "memory");
    __syncthreads();

    // ---- pass 1: row max ----
    float m = -__builtin_inff();
    for (int j = tid; j < CLS; j += TPB) m = fmaxf(m, row[j]);
    m = waveMax(m);
    if (lane == 0) wred[wave] = m;
    __syncthreads();
    float bm = wred[0];
#pragma unroll
    for (int w = 1; w < NWAVES; ++w) bm = fmaxf(bm, wred[w]);
    __syncthreads();   // before wred is reused

    // ---- pass 2: sum of exp (v_exp_f32 fast path) ----
    float se = 0.f;
    for (int j = tid; j < CLS; j += TPB) se += __expf(row[j] - bm);
    se = waveSum(se);
    if (lane == 0) wred[wave] = se;
    __syncthreads();
    float bs = wred[0];
#pragma unroll
    for (int w = 1; w < NWAVES; ++w) bs += wred[w];

    const float logZ = bm + __logf(bs);

    // soft-label term: counts are 0.1 per annotator answer (duplicates add);
    // gather the 10 answers in parallel
    if (tid < NA) {
        int a = answers[(size_t)i * NA + tid];
        tacc[tid] = weight[a] * (row[a] - logZ);
    }
    __syncthreads();

    if (tid == 0) {
        float acc = 0.f;
#pragma unroll
        for (int k = 0; k < NA; ++k) acc += tacc[k];

        int   y  = mode_answer[i];
        float wy = weight[y];

        float4 p;
        p.x = -0.1f * acc;            // per-row weighted soft-CE
        p.y = wy * (logZ - row[y]);   // w_y * nll
        p.z = wy;                     // w_y
        p.w = (y == 0) ? 1.f : 0.f;   // for rate
        partials[i] = p;
    }
}

// --------------------------------------------------------------------------
// Kernel 2: deterministic reduction of the 16384 float4 partials -> scalar.
// --------------------------------------------------------------------------
__global__ __launch_bounds__(TPB) void vqa_final_kernel(
    const float4* __restrict__ partials, float* __restrict__ out)
{
    __shared__ float4 wred[NWAVES];
    const int tid  = threadIdx.x;
    const int lane = tid & 31;
    const int wave = tid >> 5;

    float4 acc = make_float4(0.f, 0.f, 0.f, 0.f);
    for (int i = tid; i < BATCH; i += TPB) {
        float4 p = partials[i];
        acc.x += p.x; acc.y += p.y; acc.z += p.z; acc.w += p.w;
    }
    acc.x = waveSum(acc.x);
    acc.y = waveSum(acc.y);
    acc.z = waveSum(acc.z);
    acc.w = waveSum(acc.w);
    if (lane == 0) wred[wave] = acc;
    __syncthreads();

    if (tid == 0) {
        float4 t = wred[0];
#pragma unroll
        for (int w = 1; w < NWAVES; ++w) {
            t.x += wred[w].x; t.y += wred[w].y; t.z += wred[w].z; t.w += wred[w].w;
        }
        const float sumWeighted = t.x;
        const float sumWNll     = t.y;
        const float sumWy       = t.z;
        const float n0          = t.w;

        const float invB     = 1.f / (float)BATCH;
        const float rate     = (0.8f * n0 + 0.2f * ((float)BATCH - n0)) * invB;
        const float weighted = sumWeighted * invB;
        const float standard = sumWNll / sumWy;
        out[0] = rate * weighted + (1.f - rate) * standard;
    }
}

extern "C" void kernel_launch(void* const* d_in, const int* in_sizes, int n_in,
                              void* d_out, int out_size, void* d_ws, size_t ws_size,
                              hipStream_t stream) {
    (void)in_sizes; (void)n_in; (void)out_size; (void)ws_size;

    const float* pred        = (const float*)d_in[0];
    const float* weight      = (const float*)d_in[1];
    const int*   answers     = (const int*)d_in[2];
    const int*   mode_answer = (const int*)d_in[3];

    float4* partials = (float4*)d_ws;          // BATCH * 16 B = 256 KB scratch
    float*  out      = (float*)d_out;

    vqa_row_kernel<<<BATCH, TPB, 0, stream>>>(pred, weight, answers, mode_answer, partials);
    vqa_final_kernel<<<1, TPB, 0, stream>>>(partials, out);
}